// Dictionnary_89386859364949
// MI455X (gfx1250) — compile-verified
//
#include <hip/hip_runtime.h>
#include <math.h>

#define NPATCH 31752   // 8*63*63
#define NWG    497     // ceil(NPATCH/64)
#define DDIM   192
#define NA     256
#define HP     63

typedef float v2f __attribute__((ext_vector_type(2)));
typedef float v8f __attribute__((ext_vector_type(8)));
typedef unsigned int v4u __attribute__((ext_vector_type(4)));
typedef int v4i __attribute__((ext_vector_type(4)));
typedef int v8i __attribute__((ext_vector_type(8)));

__device__ __forceinline__ v8f wmma4(v2f a, v2f b, v8f c) {
  // D = A(16x4 f32) * B(4x16 f32) + C(16x16 f32)
  return __builtin_amdgcn_wmma_f32_16x16x4_f32(false, a, false, b, (short)0, c,
                                               false, false);
}

// B-operand tiled layout (float index). row = k (0..255), col = n (0..63).
// Tile (kt,nt) stored as 32 lanes x 2 floats: lane = (n%16) + 16*((k%4)>>1),
// vgpr j = k%2.  Contiguous per-lane float2 -> conflict-free LDS access.
__device__ __forceinline__ int bAddr(int row, int col) {
  int kt = row >> 2, kk = row & 3;
  int lb = (col & 15) + ((kk >> 1) << 4);
  return (((kt << 2) + (col >> 4)) * 32 + lb) * 2 + (kk & 1);
}

// A-operand tiled layout for X (row = m, col = k), 16 m-tiles x 64 k-tiles.
__device__ __forceinline__ int aAddrX(int m, int k) {
  int kt = k >> 2, kk = k & 3;
  int la = (m & 15) + ((kk >> 1) << 4);
  return ((((m >> 4) << 6) + kt) * 32 + la) * 2 + (kk & 1);
}

// ---------------------------------------------------------------- zero output
__global__ void k_zero(float* out) {
  out[blockIdx.x * 512 + threadIdx.x] = 0.0f;
}

// ------------------------------------------------- normalize atoms -> An (ws)
__global__ void k_norm(const float* __restrict__ atoms, float* __restrict__ An) {
  int m = threadIdx.x;  // 256 threads, 1 block
  float s = 0.f;
  for (int d = 0; d < DDIM; ++d) { float v = atoms[m * DDIM + d]; s += v * v; }
  float inv = 1.0f / sqrtf(s);
  for (int d = 0; d < DDIM; ++d) An[m * DDIM + d] = atoms[m * DDIM + d] * inv;
}

// ------------------------------------------- X = An An^T (row-major + tiled)
__global__ __launch_bounds__(512) void k_gram(const float* __restrict__ An,
                                              float* __restrict__ Xp,
                                              float* __restrict__ Xt) {
  const int tid = threadIdx.x, lane = tid & 31, wave = tid >> 5;
  const int h = lane >> 4, ln = lane & 15;
  for (int i = 0; i < 16; ++i) {
    int t = wave * 16 + i;
    int mt = t >> 4, nt = t & 15;
    v8f acc = {};
    for (int kt = 0; kt < 48; ++kt) {
      int ko = kt * 4 + 2 * h;
      v2f a = *(const v2f*)(An + (mt * 16 + ln) * DDIM + ko);  // An[m][k..k+1]
      v2f b = *(const v2f*)(An + (nt * 16 + ln) * DDIM + ko);  // An[n][k..k+1]
      acc = wmma4(a, b, acc);
    }
#pragma unroll
    for (int j = 0; j < 8; ++j) {
      int m = mt * 16 + j + 8 * h;
      int n = nt * 16 + ln;
      Xp[m * NA + n] = acc[j];
      Xt[aAddrX(m, n)] = acc[j];
    }
  }
}

// ---------------------------------------- power iteration: L = ||X||_2 (PSD)
__global__ void k_power(const float* __restrict__ Xp, float* __restrict__ scal) {
  __shared__ float v[NA];
  __shared__ float red[NA];
  int tid = threadIdx.x;  // 256 threads
  v[tid] = 1.0f;
  __syncthreads();
  float L = 1.0f;
  for (int it = 0; it < 96; ++it) {
    float s = 0.f;
    for (int n = 0; n < NA; ++n) s += Xp[tid * NA + n] * v[n];
    red[tid] = s * s;
    __syncthreads();
    for (int o = 128; o > 0; o >>= 1) {
      if (tid < o) red[tid] += red[tid + o];
      __syncthreads();
    }
    float nrm = sqrtf(red[0]);
    __syncthreads();
    v[tid] = s / nrm;
    L = nrm;
    __syncthreads();
  }
  if (tid == 0) { scal[0] = 1.0f / L; scal[1] = 0.1f / L; scal[2] = L; }
}

// ------------------------------------------------------------- fused ISTA
// Per WG (512 thr = 16 waves): 64 patches. LDS: X tiles 256KB + c tiles 64KB.
// Wave w owns M-tiles {2*(w>>1), +1} and N-tiles {2*(w&1), +1}.
extern __shared__ float lds[];

__global__ __launch_bounds__(512) void k_mega(const float* __restrict__ y,
                                              const float* __restrict__ An,
                                              const float* __restrict__ Xt,
                                              const float* __restrict__ scal,
                                              float* __restrict__ meanG,
                                              float* __restrict__ out) {
  float* Xl = lds;              // 65536 floats (A-operand tiles of X)
  float* Cl = lds + 65536;      // 16384 floats (B-operand tiles of p then c)
  const int tid = threadIdx.x, lane = tid & 31, wave = tid >> 5;
  const int h = lane >> 4, ln = lane & 15;
  const int pBase = blockIdx.x * 64;

#if __has_builtin(__builtin_amdgcn_tensor_load_to_lds)
  // ---- TDM: DMA the pre-swizzled 256KB X straight into LDS (offset 0).
  // One descriptor per WG, issued by wave 0; overlaps with extraction + q.
  // Tensor: 8-byte elements, 2-D 8192x4 tile == 262144 bytes.
  if (wave == 0) {
    unsigned long long ga = (unsigned long long)(uintptr_t)Xt;
    v4u g0;
    g0[0] = 1u;                                   // count=1, user mode
    g0[1] = 0u;                                   // lds_addr = 0 bytes
    g0[2] = (unsigned)(ga & 0xffffffffu);         // global_addr[31:0]
    g0[3] = (unsigned)((ga >> 32) & 0x1ffffffu) | 0x80000000u;  // type=2
    v8i g1;
    g1[0] = 0x00030000;   // wg_mask=0, data_size=3 (8B), no barrier/iter/pad
    g1[1] = 0x20000000;   // tensor_dim0[15:0]=8192 in bits[31:16]
    g1[2] = 0x00040000;   // tensor_dim0[31:16]=0, tensor_dim1=4
    g1[3] = 0x20000000;   // tile_dim0=8192 in bits[31:16]
    g1[4] = 0x00000004;   // tile_dim1=4, tile_dim2=0
    g1[5] = 8192;         // tensor_dim0_stride[31:0]
    g1[6] = 0;            // stride0[47:32]=0, stride1[15:0]=0
    g1[7] = 0;
    v4i gz4 = {};
#if __clang_major__ >= 23
    v8i gz8 = {};
    __builtin_amdgcn_tensor_load_to_lds(g0, g1, gz4, gz4, gz8, 0);
#else
    __builtin_amdgcn_tensor_load_to_lds(g0, g1, gz4, gz4, 0);
#endif
  }
#endif

  // ---- patch extraction + mean centering into B-tiled LDS ----
  {
    int pl = tid >> 3, part = tid & 7;             // 8 threads per patch
    int pidx = pBase + pl;
    int pc = (pidx < NPATCH) ? pidx : (NPATCH - 1);
    int b = pc / (HP * HP);
    int r = pc - b * (HP * HP);
    int hp = r / HP, wp = r - hp * HP;
    const float* yb = y + (size_t)b * 3 * 65536 + (hp * 4) * 256 + wp * 4;
    float vals[24];
    float s = 0.f;
#pragma unroll
    for (int i = 0; i < 24; ++i) {
      int d = part * 24 + i;
      int ch = d >> 6, dy = (d >> 3) & 7, dx = d & 7;
      float vv = yb[ch * 65536 + dy * 256 + dx];
      vals[i] = vv;
      s += vv;
    }
    float* scratch = Cl + 12288;  // 512 floats
    float* meanL = Cl + 12800;    // 64 floats
    scratch[tid] = s;
    __syncthreads();
    if (tid < 64) {
      float m = 0.f;
      for (int j = 0; j < 8; ++j) m += scratch[tid * 8 + j];
      m *= (1.0f / DDIM);
      meanL[tid] = m;
      if (pBase + tid < NPATCH) meanG[pBase + tid] = m;
    }
    __syncthreads();
    float mn = meanL[pl];
#pragma unroll
    for (int i = 0; i < 24; ++i) {
      int d = part * 24 + i;
      Cl[bAddr(d, pl)] = vals[i] - mn;
    }
  }
  __syncthreads();

#if !__has_builtin(__builtin_amdgcn_tensor_load_to_lds)
  // ---- fallback: manual coalesced copy of pre-swizzled X into LDS ----
  for (int i = 0; i < 32; ++i) {
    int off = (i * 512 + tid) * 4;
    *(float4*)(Xl + off) = *(const float4*)(Xt + off);
  }
#endif

  // ---- q = An * p^T  (K = 192), per-wave tiles kept in registers ----
  const int mt0 = (wave >> 1) * 2;
  const int nt0 = (wave & 1) * 2;
  v8f q00 = {}, q01 = {}, q10 = {}, q11 = {};
  for (int kt = 0; kt < 48; ++kt) {
    int ko = kt * 4 + 2 * h;
    v2f a0 = *(const v2f*)(An + ((mt0 + 0) * 16 + ln) * DDIM + ko);
    v2f a1 = *(const v2f*)(An + ((mt0 + 1) * 16 + ln) * DDIM + ko);
    v2f b0 = *(const v2f*)(Cl + ((kt * 4 + nt0 + 0) * 32 + lane) * 2);
    v2f b1 = *(const v2f*)(Cl + ((kt * 4 + nt0 + 1) * 32 + lane) * 2);
    q00 = wmma4(a0, b0, q00);
    q01 = wmma4(a0, b1, q01);
    q10 = wmma4(a1, b0, q10);
    q11 = wmma4(a1, b1, q11);
  }
  __syncthreads();

  // ---- c = 0 in LDS tiles ----
  for (int i = 0; i < 8; ++i) {
    int off = (i * 512 + tid) * 4;
    *(float4*)(Cl + off) = make_float4(0.f, 0.f, 0.f, 0.f);
  }

#if __has_builtin(__builtin_amdgcn_tensor_load_to_lds)
  if (wave == 0) { __builtin_amdgcn_s_wait_tensorcnt(0); }  // X fully in LDS
#endif
  __syncthreads();

  const float Linv = scal[0];
  const float thr = scal[1];
  v8f c00 = {}, c01 = {}, c10 = {}, c11 = {};

  // ---- 25 ISTA iterations, fully on-WGP ----
  for (int it = 0; it < 25; ++it) {
    v8f t00 = {}, t01 = {}, t10 = {}, t11 = {};
    for (int kt = 0; kt < 64; ++kt) {
      v2f a0 = *(const v2f*)(Xl + (((mt0 + 0) * 64 + kt) * 32 + lane) * 2);
      v2f a1 = *(const v2f*)(Xl + (((mt0 + 1) * 64 + kt) * 32 + lane) * 2);
      v2f b0 = *(const v2f*)(Cl + ((kt * 4 + nt0 + 0) * 32 + lane) * 2);
      v2f b1 = *(const v2f*)(Cl + ((kt * 4 + nt0 + 1) * 32 + lane) * 2);
      t00 = wmma4(a0, b0, t00);
      t01 = wmma4(a0, b1, t01);
      t10 = wmma4(a1, b0, t10);
      t11 = wmma4(a1, b1, t11);
    }
#pragma unroll
    for (int j = 0; j < 8; ++j) {
      float cn;
      cn = c00[j] - (t00[j] - q00[j]) * Linv;
      { float ab = fabsf(cn) - thr; c00[j] = (ab > 0.f) ? copysignf(ab, cn) : 0.f; }
      cn = c01[j] - (t01[j] - q01[j]) * Linv;
      { float ab = fabsf(cn) - thr; c01[j] = (ab > 0.f) ? copysignf(ab, cn) : 0.f; }
      cn = c10[j] - (t10[j] - q10[j]) * Linv;
      { float ab = fabsf(cn) - thr; c10[j] = (ab > 0.f) ? copysignf(ab, cn) : 0.f; }
      cn = c11[j] - (t11[j] - q11[j]) * Linv;
      { float ab = fabsf(cn) - thr; c11[j] = (ab > 0.f) ? copysignf(ab, cn) : 0.f; }
    }
    __syncthreads();  // everyone done reading old c
#pragma unroll
    for (int j = 0; j < 8; ++j) {
      int r0 = (mt0 + 0) * 16 + j + 8 * h;
      int r1 = (mt0 + 1) * 16 + j + 8 * h;
      int n0 = (nt0 + 0) * 16 + ln;
      int n1 = (nt0 + 1) * 16 + ln;
      Cl[bAddr(r0, n0)] = c00[j];
      Cl[bAddr(r0, n1)] = c01[j];
      Cl[bAddr(r1, n0)] = c10[j];
      Cl[bAddr(r1, n1)] = c11[j];
    }
    __syncthreads();  // new c visible
  }

  // ---- rec = c^T * An + mean ; overlap-add into out ----
  // wave -> patch-tile pt = wave>>2 (0..3); 3 d-tiles starting at (wave&3)*48
  const int pt = wave >> 2;
  for (int ti = 0; ti < 3; ++ti) {
    int Dbase = (wave & 3) * 48 + ti * 16;
    v8f racc = {};
    for (int kt = 0; kt < 64; ++kt) {
      int k0 = kt * 4 + 2 * h;
      int p = pt * 16 + ln;
      v2f a = *(const v2f*)(Cl + bAddr(k0, p));  // c^T[p][k0..k0+1]
      v2f bb;
      int d = Dbase + ln;
      bb.x = An[(k0 + 0) * DDIM + d];
      bb.y = An[(k0 + 1) * DDIM + d];
      racc = wmma4(a, bb, racc);
    }
#pragma unroll
    for (int j = 0; j < 8; ++j) {
      int p = pt * 16 + j + 8 * h;
      int pidx = pBase + p;
      if (pidx < NPATCH) {
        float v = racc[j] + meanG[pidx];
        int d = Dbase + ln;
        int ch = d >> 6, dy = (d >> 3) & 7, dx = d & 7;
        int b = pidx / (HP * HP);
        int r = pidx - b * (HP * HP);
        int hp = r / HP, wp = r - hp * HP;
        atomicAdd(out + (size_t)(b * 3 + ch) * 65536 + (hp * 4 + dy) * 256 +
                      wp * 4 + dx,
                  v);
      }
    }
  }
}

// --------------------------------------------- divide by analytical counts
__global__ void k_div(float* out) {
  int idx = blockIdx.x * 512 + threadIdx.x;
  int w = idx & 255;
  int hh = (idx >> 8) & 255;
  int loH = (hh >= 7) ? ((hh - 4) >> 2) : 0;
  int hiH = min(62, hh >> 2);
  int loW = (w >= 7) ? ((w - 4) >> 2) : 0;
  int hiW = min(62, w >> 2);
  float cnt = (float)((hiH - loH + 1) * (hiW - loW + 1));
  out[idx] = out[idx] / cnt;
}

extern "C" void kernel_launch(void* const* d_in, const int* in_sizes, int n_in,
                              void* d_out, int out_size, void* d_ws,
                              size_t ws_size, hipStream_t stream) {
  const float* y = (const float*)d_in[0];      // (8,3,256,256)
  const float* atoms = (const float*)d_in[1];  // (256,3,8,8)
  float* out = (float*)d_out;                  // (8,3,256,256)

  float* ws = (float*)d_ws;
  float* An = ws;                 // 49152
  float* Xp = ws + 49152;         // 65536
  float* Xt = ws + 114688;        // 65536
  float* scal = ws + 180224;      // 8
  float* meanG = ws + 180232;     // 31752

  (void)hipFuncSetAttribute((const void*)k_mega,
                            hipFuncAttributeMaxDynamicSharedMemorySize, 327680);

  k_zero<<<3072, 512, 0, stream>>>(out);
  k_norm<<<1, 256, 0, stream>>>(atoms, An);
  k_gram<<<1, 512, 0, stream>>>(An, Xp, Xt);
  k_power<<<1, 256, 0, stream>>>(Xp, scal);
  k_mega<<<NWG, 512, 327680, stream>>>(y, An, Xt, scal, meanG, out);
  k_div<<<3072, 512, 0, stream>>>(out);
}